// Vanilla_AE_separate_26731876450990
// MI455X (gfx1250) — compile-verified
//
#include <hip/hip_runtime.h>

// MoE autoencoder, grouped-GEMM via expert bucketing, fp32 WMMA (gfx1250).
// B=131072 samples, SEQ=128, ENC=32, E=8 experts. Output: (B,128) fp32.
//
// LDS weight layout is k-pair interleaved: element (k,n) lives at
// [k/2][2n + (k&1)], so a WMMA B-operand (two consecutive k for fixed n)
// is a single aligned ds_load_b64 into an even VGPR pair.

#define BQ   131072
#define SEQ_ 128
#define ENC_ 32
#define NEXP 8
#define WAVES_PER_BLOCK 8
#define ROWS_PER_BLOCK  (16 * WAVES_PER_BLOCK)    // 128 samples per block
#define BLOCKS_PER_EXPERT (BQ / ROWS_PER_BLOCK)   // 1024

#define W1P_ROW (2 * ENC_ + 2)    // 66 floats: even (8B pairs stay aligned), bank-staggered
#define W2P_ROW (2 * SEQ_ + 2)    // 258 floats
#define HS_ROW  (ENC_ + 4)        // 36 floats

typedef __attribute__((ext_vector_type(2))) float v2f;
typedef __attribute__((ext_vector_type(8))) float v8f;

#define WMMA_F32(A, Bv, C) \
  __builtin_amdgcn_wmma_f32_16x16x4_f32(false, (A), false, (Bv), (short)0, (C), false, false)

__global__ void zero_counts_kernel(int* __restrict__ counts) {
  if (threadIdx.x < NEXP) counts[threadIdx.x] = 0;
}

__global__ void bucketize_kernel(const float* __restrict__ x,
                                 int* __restrict__ counts,
                                 int* __restrict__ perm) {
  int t = blockIdx.x * blockDim.x + threadIdx.x;
  if (t >= BQ) return;
  int e = (int)x[(size_t)t * (SEQ_ + 1) + SEQ_];   // expert id stored as float col 128
  int pos = atomicAdd(&counts[e], 1);
  perm[(size_t)e * BQ + pos] = t;
}

__global__ __launch_bounds__(256) void moe_ae_kernel(
    const float* __restrict__ x,
    const float* __restrict__ W1, const float* __restrict__ b1,
    const float* __restrict__ W2, const float* __restrict__ b2,
    const int* __restrict__ counts, const int* __restrict__ perm,
    float* __restrict__ out)
{
  __shared__ float w1p[SEQ_ / 2][W1P_ROW];              // 64 x 66
  __shared__ float w2p[ENC_ / 2][W2P_ROW];              // 16 x 258
  __shared__ float hs[WAVES_PER_BLOCK][16][HS_ROW];     // per-wave H tile (16 x 36)

  const int e   = blockIdx.x / BLOCKS_PER_EXPERT;
  const int blk = blockIdx.x % BLOCKS_PER_EXPERT;
  const int cnt = counts[e];
  if (blk * ROWS_PER_BLOCK >= cnt) return;              // over-launched block: cheap exit

  const int tid = threadIdx.x;
  // Stage this expert's weights into LDS, pair-interleaved.
  for (int i = tid; i < SEQ_ * ENC_; i += 256) {
    const int k = i >> 5, n = i & 31;
    w1p[k >> 1][(n << 1) | (k & 1)] = W1[(size_t)e * SEQ_ * ENC_ + i];
  }
  for (int i = tid; i < ENC_ * SEQ_; i += 256) {
    const int k = i >> 7, n = i & 127;
    w2p[k >> 1][(n << 1) | (k & 1)] = W2[(size_t)e * ENC_ * SEQ_ + i];
  }
  __syncthreads();

  const int wave = tid >> 5;        // wave32
  const int lane = tid & 31;
  const int half = lane >> 4;       // 0: lanes 0-15, 1: lanes 16-31
  const int ml   = lane & 15;       // M for A-operand, N for B/C/D operands
  const int kk   = half << 1;       // K sub-offset within a 16x16x4 step

  const int row0 = (blk * WAVES_PER_BLOCK + wave) * 16; // this wave's tile
  if (row0 >= cnt) return;

  const int* bucket = perm + (size_t)e * BQ;

  // A-operand source row for this lane (clamped for partial tiles; stores are guarded).
  const int rA    = row0 + ml;
  const int ridxA = bucket[(rA < cnt) ? rA : row0];
  const float* xrow = x + (size_t)ridxA * (SEQ_ + 1);

  // ---------- stage 1: H(16x32) = relu( X(16x128) @ W1(128x32) + b1 ) ----------
  // Split K into two halves -> 4 independent WMMA chains for ILP.
  v8f acc0a = {}, acc0b = {};   // N = 0..15
  v8f acc1a = {}, acc1b = {};   // N = 16..31
  #pragma unroll
  for (int k0 = 0; k0 < SEQ_ / 2; k0 += 4) {
    v2f aA; aA.x = xrow[k0 + kk];      aA.y = xrow[k0 + kk + 1];
    v2f aB; aB.x = xrow[64 + k0 + kk]; aB.y = xrow[64 + k0 + kk + 1];
    const int kpA = (k0 >> 1) + half;
    const int kpB = 32 + (k0 >> 1) + half;
    const v2f bAlo = *(const v2f*)&w1p[kpA][ml << 1];
    const v2f bAhi = *(const v2f*)&w1p[kpA][(16 + ml) << 1];
    const v2f bBlo = *(const v2f*)&w1p[kpB][ml << 1];
    const v2f bBhi = *(const v2f*)&w1p[kpB][(16 + ml) << 1];
    acc0a = WMMA_F32(aA, bAlo, acc0a);
    acc1a = WMMA_F32(aA, bAhi, acc1a);
    acc0b = WMMA_F32(aB, bBlo, acc0b);
    acc1b = WMMA_F32(aB, bBhi, acc1b);
  }

  // Merge chains, add bias (per-column == per-lane scalar), ReLU, write H to LDS
  // row-major so stage-2 A operands are contiguous aligned pairs.
  const float b1lo = b1[e * ENC_ + ml];
  const float b1hi = b1[e * ENC_ + 16 + ml];
  #pragma unroll
  for (int i = 0; i < 8; ++i) {
    const int mrow = (half << 3) + i;           // C-layout row
    hs[wave][mrow][ml]      = fmaxf(acc0a[i] + acc0b[i] + b1lo, 0.f);
    hs[wave][mrow][16 + ml] = fmaxf(acc1a[i] + acc1b[i] + b1hi, 0.f);
  }
  // Same-wave LDS store->load is in-order (DScnt); compiler inserts waits.

  // Stage-2 A operands depend only on k: hoist all 8 pair-loads out of the nt loop.
  v2f ah[8];
  #pragma unroll
  for (int t = 0; t < 8; ++t)
    ah[t] = *(const v2f*)&hs[wave][ml][(t << 2) + kk];

  // Output row indices for the scatter (uniform per half-wave, hoisted).
  int  ridxO[8];
  bool validO[8];
  #pragma unroll
  for (int i = 0; i < 8; ++i) {
    const int rr = row0 + (half << 3) + i;
    validO[i] = (rr < cnt);
    ridxO[i]  = bucket[validO[i] ? rr : row0];
  }

  // ---------- stage 2: Y(16x128) = H(16x32) @ W2(32x128) + b2, scattered ----------
  #pragma unroll
  for (int nt = 0; nt < 8; ++nt) {
    const int n0 = nt << 4;
    v8f acc = {};
    #pragma unroll
    for (int t = 0; t < 8; ++t) {
      const int kp = (t << 1) + half;           // ((t*4)+kk)/2
      const v2f b = *(const v2f*)&w2p[kp][(n0 + ml) << 1];
      acc = WMMA_F32(ah[t], b, acc);
    }
    const float bb = b2[e * SEQ_ + n0 + ml];
    #pragma unroll
    for (int i = 0; i < 8; ++i) {
      if (validO[i])
        out[(size_t)ridxO[i] * SEQ_ + n0 + ml] = acc[i] + bb;
    }
  }
}

extern "C" void kernel_launch(void* const* d_in, const int* in_sizes, int n_in,
                              void* d_out, int out_size, void* d_ws, size_t ws_size,
                              hipStream_t stream) {
  const float* x  = (const float*)d_in[0];   // (B, 129)
  const float* W1 = (const float*)d_in[1];   // (8, 128, 32)
  const float* b1 = (const float*)d_in[2];   // (8, 32)
  const float* W2 = (const float*)d_in[3];   // (8, 32, 128)
  const float* b2 = (const float*)d_in[4];   // (8, 128)
  float* out = (float*)d_out;                // (B, 128)

  int* counts = (int*)d_ws;                  // 8 ints
  int* perm   = counts + 8;                  // 8 * B ints (~4 MB)

  zero_counts_kernel<<<1, 32, 0, stream>>>(counts);
  bucketize_kernel<<<BQ / 256, 256, 0, stream>>>(x, counts, perm);
  moe_ae_kernel<<<NEXP * BLOCKS_PER_EXPERT, 256, 0, stream>>>(
      x, W1, b1, W2, b2, counts, perm, out);
}